// AddNoise_54279796687281
// MI455X (gfx1250) — compile-verified
//
#include <hip/hip_runtime.h>
#include <hip/hip_bf16.h>
#include <math.h>

// ---------------------------------------------------------------------------
// Kaldi LinearResample 16000 -> 22050, LPW=6, mapped onto CDNA5 WMMA.
//   P=441 phases, stride=320, 13 taps, K padded to 24 -> 6 x wmma_f32_16x16x4_f32
//   Y[q,p] tile (16x16) = X-window (16x24) @ W-band (24x16), exact fp32 GEMM.
// ---------------------------------------------------------------------------

typedef float v2f __attribute__((ext_vector_type(2)));
typedef float v8f __attribute__((ext_vector_type(8)));

namespace rs {
constexpr int    NSAMP  = 960000;      // input samples per batch row
constexpr int    P      = 441;         // phases
constexpr int    STRIDE = 320;         // input stride per output frame block
constexpr int    NQ     = 3000;        // output frames (NQ*P == tot)
constexpr long   NT     = 1323000;     // outputs per batch row
constexpr int    NPT    = 28;          // p-tiles of 16 (28*16 >= 441)
constexpr int    NQT    = 188;         // q-tiles of 16 (188*16 >= 3000)
constexpr int    QPW    = 4;           // q-tiles per wave (188 = 47*4)
constexpr int    NAPF   = NPT * 6 * 64;          // floats in packed weight tiles
constexpr double PI_D   = 3.14159265358979323846;
}

// ---------------------------------------------------------------------------
// Setup: build 28 weight tiles (16x24, zero-padded band) directly in the
// WMMA B-operand lane layout, plus fi[p0] per tile. Double math mirrors the
// numpy reference exactly. apack index = ((pt*6+m)*32 + lane)*2 + e.
// ---------------------------------------------------------------------------
__global__ void build_filters_kernel(float* __restrict__ apack,
                                     int* __restrict__ fiArr)
{
    const int gid = blockIdx.x * blockDim.x + threadIdx.x;
    const double ORIGd  = 16000.0;
    const double NEWd   = 22050.0;
    const double cutoff = 0.99 * 0.5 * 16000.0;          // 7920
    const double ww     = 6.0 / (2.0 * cutoff);

    if (gid < rs::NPT) {
        double out_t0 = (double)(gid * 16) / NEWd;
        fiArr[gid] = (int)ceil((out_t0 - ww) * ORIGd);
    }
    if (gid >= rs::NAPF) return;

    int pt   = gid / 384;          // p-tile
    int rem  = gid % 384;
    int m    = rem / 64;           // k-chunk (0..5)
    int r2   = rem % 64;
    int lane = r2 >> 1;
    int e    = r2 & 1;
    int ploc = lane & 15;
    int kk   = (lane >> 4) << 1;   // 0 for lanes 0-15, 2 for lanes 16-31
    int r    = 4 * m + kk + e;     // K index 0..23
    int p    = pt * 16 + ploc;

    float val = 0.0f;
    if (p < rs::P) {
        double out_t  = (double)p / NEWd;
        double fi_p   = ceil((out_t - ww) * ORIGd);
        double out_t0 = (double)(pt * 16) / NEWd;
        double fi_0   = ceil((out_t0 - ww) * ORIGd);
        int off = (int)(fi_p - fi_0);
        int j   = r - off;
        if (j >= 0 && j < 13) {
            double dt = (fi_p + (double)j) / ORIGd - out_t;
            if (fabs(dt) < ww) {
                double wv = 0.5 * (1.0 + cos(2.0 * rs::PI_D * cutoff / 6.0 * dt));
                if (dt != 0.0) wv *= sin(2.0 * rs::PI_D * cutoff * dt) / (rs::PI_D * dt);
                else           wv *= 2.0 * cutoff;
                wv /= ORIGd;
                val = (float)wv;
            }
        }
    }
    apack[gid] = val;
}

// ---------------------------------------------------------------------------
// Main kernel: one wave per (batch, p-tile, group-of-4 q-tiles).
//   A operand = input windows (rows = q frames), B operand = weight band.
//   D rows = q, cols = p -> 16 consecutive output floats per D register row.
// ---------------------------------------------------------------------------
__global__ __launch_bounds__(32)
void resample_wmma_kernel(const float* __restrict__ x,
                          float* __restrict__ y,
                          const float2* __restrict__ apack2,
                          const int* __restrict__ fiArr)
{
    const int lane = threadIdx.x;           // 0..31, wave32
    const int pt   = blockIdx.y;
    const int b    = blockIdx.z;
    const int pl   = lane & 15;             // p_loc (B col) and q_loc (A row)
    const int kk   = (lane >> 4) << 1;      // K sub-offset per half-wave
    const int p0   = pt * 16;
    const int fi0  = fiArr[pt];

    // ---- weight tile (B operand), reused across the 4 q-tiles ----
    v2f wb[6];
    const float2* ap = apack2 + (size_t)(pt * 6) * 32 + lane;
#pragma unroll
    for (int m = 0; m < 6; ++m) {
        float2 t = ap[m * 32];
        v2f v; v.x = t.x; v.y = t.y;
        wb[m] = v;
    }

    const float* xb = x + (size_t)b * rs::NSAMP;
    const int p = p0 + pl;

#pragma unroll
    for (int s = 0; s < rs::QPW; ++s) {
        const int qt = blockIdx.x * rs::QPW + s;
        const int q0 = qt * 16;

        // ---- input window tile (A operand): row = q_loc = pl ----
        const int q    = q0 + pl;
        const int base = fi0 + q * rs::STRIDE;   // first K index for this row
        v2f ad[6];
        if (base >= 0 && base + 23 < rs::NSAMP) {
#pragma unroll
            for (int m = 0; m < 6; ++m) {
                const float* src = xb + base + 4 * m + kk;
                v2f v; v.x = src[0]; v.y = src[1];   // scalar b32 loads: no alignment hazard
                ad[m] = v;
            }
        } else {
#pragma unroll
            for (int m = 0; m < 6; ++m) {
                int i0 = base + 4 * m + kk;
                int i1 = i0 + 1;
                v2f v;
                v.x = (i0 >= 0 && i0 < rs::NSAMP) ? xb[i0] : 0.0f;
                v.y = (i1 >= 0 && i1 < rs::NSAMP) ? xb[i1] : 0.0f;
                ad[m] = v;
            }
        }

        // ---- 16x16x24 fp32 GEMM via 6 chained WMMA ----
        v8f c = {};
#pragma unroll
        for (int m = 0; m < 6; ++m) {
            c = __builtin_amdgcn_wmma_f32_16x16x4_f32(
                    /*neg_a=*/false, ad[m],
                    /*neg_b=*/false, wb[m],
                    /*c_mod=*/(short)0, c,
                    /*reuse_a=*/false, /*reuse_b=*/false);
        }

        // ---- store: D vgpr v -> row q_loc = v (+8 for upper half), col p_loc = pl
        if (p < rs::P) {
            const int qrow0 = q0 + ((lane < 16) ? 0 : 8);
            float* dst = y + (size_t)b * rs::NT + (size_t)qrow0 * rs::P + p;
            if (q0 + 16 <= rs::NQ) {
                // wave-uniform interior case: 8 unguarded stores (clause-able);
                // each wave-level store covers two fully-dense 64B output rows.
#pragma unroll
                for (int v = 0; v < 8; ++v) {
                    dst[(size_t)v * rs::P] = c[v];
                }
            } else {
                // only the final q-tile (q0 == 2992) lands here
#pragma unroll
                for (int v = 0; v < 8; ++v) {
                    if (qrow0 + v < rs::NQ) dst[(size_t)v * rs::P] = c[v];
                }
            }
        }
    }
}

// ---------------------------------------------------------------------------
extern "C" void kernel_launch(void* const* d_in, const int* in_sizes, int n_in,
                              void* d_out, int out_size, void* d_ws, size_t ws_size,
                              hipStream_t stream) {
    const float* x = (const float*)d_in[0];
    float* y = (float*)d_out;

    // workspace: packed weight tiles (43,008 B) then fi[28] (112 B)
    float* apack = (float*)d_ws;
    int*   fiArr = (int*)((char*)d_ws + (size_t)rs::NAPF * sizeof(float));

    build_filters_kernel<<<(rs::NAPF + 127) / 128, 128, 0, stream>>>(apack, fiArr);

    dim3 grid(rs::NQT / rs::QPW, rs::NPT, 8);   // (47, 28, 8), one wave per block
    resample_wmma_kernel<<<grid, 32, 0, stream>>>(x, y, (const float2*)apack, fiArr);
}